// Token2Vec_26714696581183
// MI455X (gfx1250) — compile-verified
//
#include <hip/hip_runtime.h>

// Problem constants (match reference).
#define B_  8
#define T_  2048
#define H_  1024
#define W_  1024

typedef float v2f __attribute__((ext_vector_type(2)));
typedef float v4f __attribute__((ext_vector_type(4)));
typedef float v8f __attribute__((ext_vector_type(8)));

// context (B,W,H) floats, then mask (B,W) floats
#define MASK_OFF ((size_t)B_ * W_ * H_)

#if defined(__gfx1250__) && __has_builtin(__builtin_amdgcn_wmma_f32_16x16x4_f32)
#define USE_WMMA 1
#else
#define USE_WMMA 0
#endif

// One block per output position (b, i). 256 threads = 8 waves; each wave owns a
// 128-wide slice of H, split into 8 WMMA tiles of 16 h-values.
__global__ __launch_bounds__(256) void token2vec_span_kernel(
    const float* __restrict__ hidden,    // (B,T,H,4)
    const int*   __restrict__ word_ids,  // (B,T) sorted along T
    const int*   __restrict__ span,      // (B,2)
    float*       __restrict__ out)
{
    const int p    = blockIdx.x;          // b*W + i
    const int b    = p >> 10;             // / W_
    const int i    = p & (W_ - 1);
    const int tid  = threadIdx.x;
    const int lane = tid & 31;
    const int wave = tid >> 5;

    __shared__ int s_lo, s_hi;

    const int start = span[2 * b];
    const int end   = span[2 * b + 1];
    const int wid   = start + i;
    const bool maskv = (wid < end);
    int widc = wid;
    if (widc < 0) widc = 0;
    if (widc > W_ - 1) widc = W_ - 1;

    // Token range for word `widc` via binary search on the sorted row.
    if (tid == 0) {
        const int* wrow = word_ids + b * T_;
        int lo = 0, n = T_;
        while (n > 0) {                       // first idx with wrow[idx] >= widc
            int half = n >> 1, mid = lo + half;
            if (wrow[mid] < widc) { lo = mid + 1; n -= half + 1; }
            else n = half;
        }
        int hi = 0, m = T_;
        while (m > 0) {                       // first idx with wrow[idx] > widc
            int half = m >> 1, mid = hi + half;
            if (wrow[mid] <= widc) { hi = mid + 1; m -= half + 1; }
            else m = half;
        }
        s_lo = lo; s_hi = hi;
    }
    __syncthreads();

    const size_t rowbase = (size_t)p * H_;

    if (!maskv) {                             // masked row: zeros (block-uniform branch)
        v4f z = {0.f, 0.f, 0.f, 0.f};
        *(v4f*)(out + rowbase + 4 * tid) = z; // 256 threads * 4 = H
        if (tid == 0) out[MASK_OFF + p] = 0.0f;
        return;
    }

    const int lo  = s_lo, hi = s_hi;
    const int cnt = hi - lo;
    const float scale = 1.0f / (float)(cnt > 0 ? cnt : 1);

    const int h0w  = wave * 128;                                 // wave's H slice
    const int aoff = (lane & 15) * 4 + (lane >> 4) * 2;          // f32 16x4 A layout:
                                                                 // lanes 0-15: l=0,1; lanes 16-31: l=2,3

#if USE_WMMA
    v8f acc[8];
#pragma unroll
    for (int j = 0; j < 8; ++j) acc[j] = {};
    const v2f bmat = {0.25f, 0.25f};          // uniform B => layout-independent

    for (int t = lo; t < hi; ++t) {
        const float* tok = hidden + ((size_t)(b * T_ + t) * H_) * 4;
        if (t + 1 < hi)                       // -> global_prefetch_b8
            __builtin_prefetch(hidden + ((size_t)(b * T_ + t + 1) * H_) * 4
                               + (size_t)h0w * 4 + aoff, 0, 1);
#pragma unroll
        for (int j = 0; j < 8; ++j) {
            v2f a = *(const v2f*)(tok + (size_t)(h0w + j * 16) * 4 + aoff);
            // D(16x16) = A(16x4) x 0.25 + C : every column = per-row means; C chains token sum
            acc[j] = __builtin_amdgcn_wmma_f32_16x16x4_f32(
                false, a, false, bmat, (short)0, acc[j], false, false);
        }
    }

    // D layout: VGPR r -> lanes 0-15 hold M=r, lanes 16-31 hold M=r+8 (all columns equal).
    const bool active = (lane & 8) == 0;      // 16 unique lanes per tile
    const int  r  = lane & 7;
    const int  hl = r + ((lane >> 4) << 3);   // lanes 0-7 -> h 0-7, lanes 16-23 -> h 8-15
#pragma unroll
    for (int j = 0; j < 8; ++j) {
        v8f d = acc[j];
        float v = (r < 4) ? ((r < 2) ? (r == 0 ? d[0] : d[1]) : (r == 2 ? d[2] : d[3]))
                          : ((r < 6) ? (r == 4 ? d[4] : d[5]) : (r == 6 ? d[6] : d[7]));
        if (active) out[rowbase + h0w + j * 16 + hl] = v * scale;
    }
#else
    // VALU fallback: per-lane partial (l0+l1 or l2+l3), combine halves via lane-xor-16.
    float accs[8];
#pragma unroll
    for (int j = 0; j < 8; ++j) accs[j] = 0.0f;
    for (int t = lo; t < hi; ++t) {
        const float* tok = hidden + ((size_t)(b * T_ + t) * H_) * 4;
#pragma unroll
        for (int j = 0; j < 8; ++j) {
            v2f a = *(const v2f*)(tok + (size_t)(h0w + j * 16) * 4 + aoff);
            accs[j] += a.x + a.y;
        }
    }
#pragma unroll
    for (int j = 0; j < 8; ++j) {
        float s = accs[j] + __shfl_xor(accs[j], 16, 32);
        if (lane < 16) out[rowbase + h0w + j * 16 + lane] = s * 0.25f * scale;
    }
#endif

    if (tid == 0) out[MASK_OFF + p] = 1.0f;
}

extern "C" void kernel_launch(void* const* d_in, const int* in_sizes, int n_in,
                              void* d_out, int out_size, void* d_ws, size_t ws_size,
                              hipStream_t stream) {
    (void)in_sizes; (void)n_in; (void)d_ws; (void)ws_size; (void)out_size;
    const float* hidden   = (const float*)d_in[0];
    const int*   word_ids = (const int*)d_in[1];
    const int*   span     = (const int*)d_in[2];
    float*       out      = (float*)d_out;

    token2vec_span_kernel<<<B_ * W_, 256, 0, stream>>>(hidden, word_ids, span, out);
}